// SelfAttentionPooling_49246095016345
// MI455X (gfx1250) — compile-verified
//
#include <hip/hip_runtime.h>
#include <stdint.h>

// ---------------- CDNA5 feature detection (also keeps host pass parseable) ----
#ifndef __has_builtin
#define __has_builtin(x) 0
#endif
#if __has_builtin(__builtin_amdgcn_wmma_f32_16x16x4_f32)
#define HAVE_WMMA_F32 1
#else
#define HAVE_WMMA_F32 0
#endif
#if __has_builtin(__builtin_amdgcn_tensor_load_to_lds) && __has_builtin(__builtin_amdgcn_s_wait_tensorcnt)
#define HAVE_TDM 1
#else
#define HAVE_TDM 0
#endif

typedef float        v2f   __attribute__((ext_vector_type(2)));
typedef float        v8f   __attribute__((ext_vector_type(8)));
typedef unsigned int u32x4 __attribute__((ext_vector_type(4)));
typedef int          i32x4 __attribute__((ext_vector_type(4)));
typedef int          i32x8 __attribute__((ext_vector_type(8)));

#define MIN_NORM   1e-15f
#define BALL_EPS   4e-3f
#define MAXNORM    (1.0f - BALL_EPS)     // c = 1
#define ATANH_CLIP (1.0f - 1e-7f)

__device__ __forceinline__ unsigned umin32(unsigned a, unsigned b) { return a < b ? a : b; }
__device__ __forceinline__ int      imin32(int a, int b)           { return a < b ? a : b; }

// logmap0 row scale: atanh(clip(n)) / max(n, MIN_NORM)   (sc = sqrt(c) = 1)
__device__ __forceinline__ float logmap0_scale(float n) {
  float nc = fmaxf(n, MIN_NORM);
  float t  = fminf(nc, ATANH_CLIP);
  return atanhf(t) / nc;
}

__device__ __forceinline__ void atomicAddF32(float* p, float v) {
#if defined(__HIP_DEVICE_COMPILE__)
  unsafeAtomicAdd(p, v);   // global_atomic_add_f32 (L2-resident 400KB accumulator)
#else
  (void)p; (void)v;
#endif
}

// ============================================================================
// K0: per-call re-init (harness does not re-poison between replays)
// ============================================================================
__global__ void k0_init(float* __restrict__ agg, unsigned* __restrict__ hist,
                        unsigned* __restrict__ state, int N, unsigned kkeep) {
  int i = blockIdx.x * blockDim.x + threadIdx.x;
  if (i < N) agg[i] = 0.0f;
  if (i < 256) hist[i] = 0u;
  if (i == 0) { state[0] = 0u; state[1] = kkeep; }
}

// ============================================================================
// K1: per-row  dot(x,w)  via V_WMMA_F32_16X16X4_F32  +  ||x||^2 via VALU,
//     then the scalar hyperbolic chain -> support[row], nrm[row].
// One wave handles a 16-row tile. Per 8-K chunk: one unconditional b128
// load/lane (row index clamped -- garbage only lands in D-rows that are never
// stored), one half-wave float2 exchange (shfl_xor 16), two chained fp32 WMMAs.
// B fragments come from a dual-region LDS: weights at [0,D), zeros at [D,2D);
// lanes with column!=0 point at the zero page, so no per-iteration masking ALU.
// A 16x4 layout: lanes 0-15 = M, vgpr{0,1}=K{0,1}; lanes 16-31 vgpr{0,1}=K{2,3}.
// C 16x16: D[m][0] lives in c[m] of lane 0 (m<8) / lane 16 (m>=8).
// ============================================================================
__global__ void k1_rowstats(const float* __restrict__ x, const float* __restrict__ w,
                            const float* __restrict__ bias, int N, int D,
                            float* __restrict__ support, float* __restrict__ nrm) {
  extern __shared__ float lw[];                    // [0,D): weight, [D,2D): zeros
  for (int t = threadIdx.x; t < 2 * D; t += blockDim.x)
    lw[t] = (t < D) ? w[t] : 0.0f;
  __syncthreads();

  const int lane = threadIdx.x & 31;
  const int wv   = threadIdx.x >> 5;
  const int tile = blockIdx.x * (blockDim.x >> 5) + wv;
  const int r0   = tile * 16;
  if (r0 >= N) return;

  const int  m     = lane & 15;
  const int  half  = lane >> 4;                    // 0: K%4 in {0,1}; 1: {2,3}
  const int  row   = r0 + m;
  const bool valid = row < N;
  const float* rp  = x + (size_t)imin32(row, N - 1) * D;   // clamped: branchless loads
  const float* bw  = lw + ((m == 0) ? 0 : D);              // zero page for columns != 0

  float n2 = 0.0f;
#if HAVE_WMMA_F32
  v8f c = {0.f,0.f,0.f,0.f,0.f,0.f,0.f,0.f};
#else
  float dp = 0.0f;
#endif

  for (int k = 0; k < D; k += 8) {
    // lanes 0-15 load cols k..k+3 of their row; lanes 16-31 load k+4..k+7
    float4 f = *(const float4*)(rp + k + half * 4);
    n2 += f.x*f.x + f.y*f.y + f.z*f.z + f.w*f.w;
#if HAVE_WMMA_F32
    // swap halves: lower sends (z,w)->upper's step-1 frag, upper sends (x,y)->lower's step-2 frag
    float sx = half ? f.x : f.z;
    float sy = half ? f.y : f.w;
    float rx = __shfl_xor(sx, 16, 32);
    float ry = __shfl_xor(sy, 16, 32);
    v2f a1, a2;
    if (half == 0) { a1 = (v2f){f.x, f.y}; a2 = (v2f){rx, ry}; }
    else           { a1 = (v2f){rx, ry};   a2 = (v2f){f.z, f.w}; }
    int kb = k + half * 2;
    v2f b1 = (v2f){ bw[kb],     bw[kb + 1] };
    v2f b2 = (v2f){ bw[kb + 4], bw[kb + 5] };
    c = __builtin_amdgcn_wmma_f32_16x16x4_f32(false, a1, false, b1, (short)0, c, false, false);
    c = __builtin_amdgcn_wmma_f32_16x16x4_f32(false, a2, false, b2, (short)0, c, false, false);
#else
    const float4 wc = *(const float4*)(lw + k + half * 4);
    dp += f.x*wc.x + f.y*wc.y + f.z*wc.z + f.w*wc.w;
#endif
  }

  float n2tot = n2 + __shfl_xor(n2, 16, 32);       // row m full ||x||^2 on lanes m, m+16

  float dot = 0.0f;
#if HAVE_WMMA_F32
#pragma unroll
  for (int j = 0; j < 8; ++j) {                    // broadcast D[m][0] to lane m
    float d0 = __shfl(c[j], 0, 32);
    float d1 = __shfl(c[j], 16, 32);
    if (lane == j)     dot = d0;
    if (lane == j + 8) dot = d1;
  }
#else
  dot = dp + __shfl_xor(dp, 16, 32);
#endif

  if (half == 0 && valid) {
    float n  = sqrtf(n2tot);
    float sv = logmap0_scale(n) * dot;             // support = s(||x||) * (x.w)
    // hyp_bias = proj(expmap0(bias)) (1-D)
    float hb = bias[0];
    float nh = fmaxf(fabsf(hb), MIN_NORM);
    float h  = tanhf(nh) * hb / nh;
    if (fabsf(h) > MAXNORM) h = copysignf(MAXNORM, h);
    // mobius_add(sv, h) in 1-D, then proj
    float x2 = sv * sv, y2 = h * h, xy = sv * h;
    float num = (1.f + 2.f * xy + y2) * sv + (1.f - x2) * h;
    float den = fmaxf(1.f + 2.f * xy + x2 * y2, MIN_NORM);
    float v  = num / den;
    float nv = fmaxf(fabsf(v), MIN_NORM);
    if (nv > MAXNORM) v = v / nv * MAXNORM;
    support[row] = v;
    nrm[row]     = n;                              // sqrt(c)*||x||, c=1
  }
}

// ============================================================================
// K2: agg[rows[e]] += vals[e] * support[cols[e]]   (agg lives in L2: 400KB)
// ============================================================================
__global__ void k2_edges(const float* __restrict__ vals, const int* __restrict__ er,
                         const int* __restrict__ ec, const float* __restrict__ support,
                         float* __restrict__ agg, int E) {
  int i = blockIdx.x * blockDim.x + threadIdx.x;
  int stride = gridDim.x * blockDim.x;
  for (; i < E; i += stride) {
    float g = vals[i] * support[ec[i]];
    atomicAddF32(&agg[er[i]], g);
  }
}

// ============================================================================
// K3: attn = w1*tanh(proj(expmap0(agg))) + (1-w1)*(1-nrm) + b ; orderable key
// ============================================================================
__global__ void k3_attn(const float* __restrict__ agg, const float* __restrict__ nrm,
                        const float* __restrict__ w1p, const float* __restrict__ bp,
                        float* __restrict__ attn_out, unsigned* __restrict__ keys, int N) {
  int i = blockIdx.x * blockDim.x + threadIdx.x;
  int stride = gridDim.x * blockDim.x;
  float w1 = w1p[0], b = bp[0];
  for (; i < N; i += stride) {
    float a  = agg[i];
    float n  = fmaxf(fabsf(a), MIN_NORM);
    float sc = tanhf(n) / n * a;                   // expmap0 in 1-D
    float nn = fmaxf(fabsf(sc), MIN_NORM);
    if (nn > MAXNORM) sc = sc / nn * MAXNORM;      // proj
    float v = w1 * tanhf(sc) + (1.f - w1) * (1.f - nrm[i]) + b;
    attn_out[i] = v;
    unsigned u = __float_as_uint(v);               // order-preserving key (desc = larger key)
    keys[i] = (u & 0x80000000u) ? ~u : (u | 0x80000000u);
  }
}

// ============================================================================
// K4: radix-select k-th largest key (4 x 8-bit passes). state={prefix, remaining}
// ============================================================================
__global__ void k4_hist(const unsigned* __restrict__ keys, int N,
                        const unsigned* __restrict__ state, unsigned* __restrict__ hist,
                        int pass) {
  __shared__ unsigned h[256];
  if (threadIdx.x < 256) h[threadIdx.x] = 0u;
  __syncthreads();
  unsigned prefix = state[0];
  int shift = 24 - 8 * pass;
  int i = blockIdx.x * blockDim.x + threadIdx.x;
  int stride = gridDim.x * blockDim.x;
  for (; i < N; i += stride) {
    unsigned key = keys[i];
    bool match = (pass == 0) || ((key >> (shift + 8)) == prefix);
    if (match) atomicAdd(&h[(key >> shift) & 255u], 1u);
  }
  __syncthreads();
  if (threadIdx.x < 256 && h[threadIdx.x]) atomicAdd(&hist[threadIdx.x], h[threadIdx.x]);
}

__global__ void k4_pick(unsigned* __restrict__ state, unsigned* __restrict__ hist) {
  if (threadIdx.x == 0) {
    unsigned r = state[1], cum = 0u, chosen = 0u;
    for (int bb = 255; bb >= 0; --bb) {
      unsigned c = hist[bb];
      if (cum + c >= r) { chosen = (unsigned)bb; break; }
      cum += c;
    }
    state[0] = (state[0] << 8) | chosen;
    state[1] = r - cum;                            // after pass 3: #ties to take = state[1]
  }
  __syncthreads();
  hist[threadIdx.x] = 0u;                          // reset for next pass
}

// ============================================================================
// K5: stable compaction of selected indices in ascending order.
//   selected = key>T, or key==T with equal-rank (by index) < need_eq
//   pos(i)   = #greater<i + min(#equal<i, need_eq)
// ============================================================================
__global__ void k5a_count(const unsigned* __restrict__ keys, int N,
                          const unsigned* __restrict__ state,
                          unsigned* __restrict__ cntG, unsigned* __restrict__ cntE) {
  __shared__ unsigned sg, se;
  if (threadIdx.x == 0) { sg = 0u; se = 0u; }
  __syncthreads();
  int i = blockIdx.x * blockDim.x + threadIdx.x;
  unsigned T = state[0];
  if (i < N) {
    unsigned key = keys[i];
    if (key > T) atomicAdd(&sg, 1u);
    else if (key == T) atomicAdd(&se, 1u);
  }
  __syncthreads();
  if (threadIdx.x == 0) { cntG[blockIdx.x] = sg; cntE[blockIdx.x] = se; }
}

__global__ void k5b_scan(unsigned* __restrict__ cntG, unsigned* __restrict__ cntE, int nb) {
  if (blockIdx.x == 0 && threadIdx.x == 0) {
    unsigned ag = 0u, ae = 0u;
    for (int bb = 0; bb < nb; ++bb) {
      unsigned g = cntG[bb], e = cntE[bb];
      cntG[bb] = ag; cntE[bb] = ae;
      ag += g; ae += e;
    }
  }
}

__global__ void k5c_emit(const unsigned* __restrict__ keys, const float* __restrict__ nrm,
                         const float* __restrict__ attn, int N,
                         const unsigned* __restrict__ state,
                         const unsigned* __restrict__ offG, const unsigned* __restrict__ offE,
                         int* __restrict__ sel, float* __restrict__ scl) {
  __shared__ unsigned wgc[8], wec[8];              // blockDim.x == 256 -> 8 waves
  int i    = blockIdx.x * blockDim.x + threadIdx.x;
  int lane = threadIdx.x & 31, wid = threadIdx.x >> 5;
  unsigned T = state[0], need = state[1];
  bool isG = false, isE = false;
  if (i < N) { unsigned key = keys[i]; isG = key > T; isE = key == T; }
  unsigned mg = (unsigned)__ballot(isG);
  unsigned me = (unsigned)__ballot(isE);
  if (lane == 0) { wgc[wid] = __popc(mg); wec[wid] = __popc(me); }
  __syncthreads();
  unsigned lt = (lane == 0) ? 0u : ((1u << lane) - 1u);
  unsigned gB = 0u, eB = 0u;
  for (int wj = 0; wj < wid; ++wj) { gB += wgc[wj]; eB += wec[wj]; }
  gB += __popc(mg & lt);
  eB += __popc(me & lt);
  unsigned gGlob = offG[blockIdx.x] + gB;          // #greater strictly before i
  unsigned eGlob = offE[blockIdx.x] + eB;          // #equal   strictly before i
  bool take = isG || (isE && eGlob < need);
  if (take) {
    unsigned pos = gGlob + umin32(eGlob, need);
    sel[pos] = i;
    scl[pos] = logmap0_scale(nrm[i]) * attn[i];    // hidden-row scale
  }
}

// ============================================================================
// K6: hidden[r] = x[sel[r]] * scl[r].
// Tensor Data Mover gather mode: one wave issues a D# gathering 8 ascending
// row indices (32-bit) straight into its LDS slab, s_wait_tensorcnt 0, then
// lanes scale from LDS and store coalesced b128 to d_out.
// ============================================================================
#define K6_ROWS_PER_WAVE 8
#define K6_WAVES 4
__global__ void k6_gather(const float* __restrict__ x, const int* __restrict__ sel,
                          const float* __restrict__ scl, float* __restrict__ out,
                          int kkeep, int N, int D) {
  extern __shared__ float lds[];                   // K6_WAVES * 8 * D floats
  const int wid  = threadIdx.x >> 5;
  const int lane = threadIdx.x & 31;
  const int base = (blockIdx.x * K6_WAVES + wid) * K6_ROWS_PER_WAVE;
  if (base >= kkeep) return;
  const int nrows = (kkeep - base < K6_ROWS_PER_WAVE) ? (kkeep - base) : K6_ROWS_PER_WAVE;
  float* myLds = lds + (size_t)wid * K6_ROWS_PER_WAVE * D;

  int idx[K6_ROWS_PER_WAVE];
#pragma unroll
  for (int j = 0; j < K6_ROWS_PER_WAVE; ++j) {
    int jj = j < nrows ? j : nrows - 1;            // pad with last index (non-decreasing)
    idx[j] = sel[base + jj];
  }

#if HAVE_TDM
  // ----- Tensor DMA descriptor (gather mode, 32-bit indices, 4-byte elems) ---
  uint32_t ldsOff = (uint32_t)(uintptr_t)myLds;    // LDS byte offset
  uint64_t ga     = (uint64_t)(uintptr_t)x;
  u32x4 g0;
  g0[0] = 1u | (1u << 30) | (1u << 31);            // count=1, 32b idx, gather_mode
  g0[1] = ldsOff;                                  // lds_addr
  g0[2] = (uint32_t)ga;                            // global_addr[31:0]
  g0[3] = (uint32_t)((ga >> 32) & 0x01FFFFFFu) | (2u << 30);  // addr[56:32] | type=2
  i32x8 g1;
  g1[0] = (int)(2u << 16);                         // data_size=2 (4B), no flags
  g1[1] = (int)(((uint32_t)D & 0xFFFFu) << 16);    // tensor_dim0[15:0] -> bits[31:16]
  g1[2] = (int)((((uint32_t)D >> 16) & 0xFFFFu) | (((uint32_t)N & 0xFFFFu) << 16)); // dim0 hi | dim1 lo
  g1[3] = (int)((((uint32_t)N >> 16) & 0xFFFFu) | (((uint32_t)D & 0xFFFFu) << 16)); // dim1 hi | tile_dim0
  g1[4] = (int)(uint32_t)nrows;                    // tile_dim1 = #valid indices
  g1[5] = (int)(uint32_t)D;                        // tensor_dim0_stride[31:0]
  g1[6] = 0;                                       // stride hi | dim1_stride lo (ignored)
  g1[7] = 0;
  i32x4 g2 = { idx[0], idx[1], idx[2], idx[3] };
  i32x4 g3 = { idx[4], idx[5], idx[6], idx[7] };
#if __clang_major__ >= 23
  i32x8 g4 = { 0, 0, 0, 0, 0, 0, 0, 0 };
  __builtin_amdgcn_tensor_load_to_lds(g0, g1, g2, g3, g4, 0);
#else
  __builtin_amdgcn_tensor_load_to_lds(g0, g1, g2, g3, 0);
#endif
  __builtin_amdgcn_s_wait_tensorcnt(0);            // wave's own TDM op complete
  for (int j = 0; j < nrows; ++j) {
    float s = scl[base + j];
    const float4* src = (const float4*)(myLds + (size_t)j * D);
    float4*       dst = (float4*)(out + (size_t)(base + j) * D);
    for (int c4 = lane; c4 * 4 < D; c4 += 32) {
      float4 v = src[c4];                          // ds_load_b128
      v.x *= s; v.y *= s; v.z *= s; v.w *= s;
      dst[c4] = v;                                 // coalesced global b128
    }
  }
#else
  for (int j = 0; j < nrows; ++j) {
    float s = scl[base + j];
    const float4* src = (const float4*)(x + (size_t)idx[j] * D);
    float4*       dst = (float4*)(out + (size_t)(base + j) * D);
    for (int c4 = lane; c4 * 4 < D; c4 += 32) {
      float4 v = src[c4];
      v.x *= s; v.y *= s; v.z *= s; v.w *= s;
      dst[c4] = v;
    }
  }
#endif
}

// ============================================================================
// Host side
// ============================================================================
extern "C" void kernel_launch(void* const* d_in, const int* in_sizes, int n_in,
                              void* d_out, int out_size, void* d_ws, size_t ws_size,
                              hipStream_t stream) {
  const float* x    = (const float*)d_in[0];
  const float* adjv = (const float*)d_in[1];
  const float* w    = (const float*)d_in[2];
  const float* bias = (const float*)d_in[3];
  const float* w1   = (const float*)d_in[4];
  const float* b    = (const float*)d_in[5];
  const int*   er   = (const int*)d_in[6];
  const int*   ec   = (const int*)d_in[7];

  const int D = in_sizes[2];                       // weight is [D,1]
  const int N = in_sizes[0] / D;
  const int E = in_sizes[1];
  const int k = N / 2;                             // int(KEEP_RATIO * N)

  float* outHidden = (float*)d_out;                // [k, D]
  float* outAttn   = outHidden + (size_t)k * D;    // [N]

  // ---- workspace carve (256B aligned) ----
  char* ws = (char*)d_ws;
  auto carve = [&](size_t bytes) {
    char* p = ws; ws += (bytes + 255) & ~(size_t)255; return p;
  };
  const int nb = (N + 255) / 256;
  float*    support = (float*)carve((size_t)N * 4);
  float*    nrm     = (float*)carve((size_t)N * 4);
  float*    agg     = (float*)carve((size_t)N * 4);
  unsigned* keys    = (unsigned*)carve((size_t)N * 4);
  unsigned* cntG    = (unsigned*)carve((size_t)nb * 4);
  unsigned* cntE    = (unsigned*)carve((size_t)nb * 4);
  int*      sel     = (int*)carve((size_t)k * 4);
  float*    scl     = (float*)carve((size_t)k * 4);
  unsigned* hist    = (unsigned*)carve(256 * 4);
  unsigned* state   = (unsigned*)carve(16);

  // K0: init accumulators / selection state
  k0_init<<<nb, 256, 0, stream>>>(agg, hist, state, N, (unsigned)k);

  // K1: WMMA row stats (8 wave-tiles of 16 rows per 256-thread block)
  {
    int tiles  = (N + 15) / 16;
    int blocks = (tiles + 7) / 8;
    k1_rowstats<<<blocks, 256, (size_t)2 * D * 4, stream>>>(x, w, bias, N, D, support, nrm);
  }

  // K2: sparse aggregate via FP32 global atomics
  {
    int blocks = (E + 255) / 256;
    if (blocks > 16384) blocks = 16384;
    k2_edges<<<blocks, 256, 0, stream>>>(adjv, er, ec, support, agg, E);
  }

  // K3: attention scores + orderable keys (attn written straight to output)
  {
    int blocks = nb > 2048 ? 2048 : nb;
    k3_attn<<<blocks, 256, 0, stream>>>(agg, nrm, w1, b, outAttn, keys, N);
  }

  // K4: 4-pass radix select of the k-th largest key
  for (int pass = 0; pass < 4; ++pass) {
    int blocks = nb > 1024 ? 1024 : nb;
    k4_hist<<<blocks, 256, 0, stream>>>(keys, N, state, hist, pass);
    k4_pick<<<1, 256, 0, stream>>>(state, hist);
  }

  // K5: stable ascending compaction of selected indices
  k5a_count<<<nb, 256, 0, stream>>>(keys, N, state, cntG, cntE);
  k5b_scan<<<1, 32, 0, stream>>>(cntG, cntE, nb);
  k5c_emit<<<nb, 256, 0, stream>>>(keys, nrm, outAttn, N, state, cntG, cntE, sel, scl);

  // K6: TDM gather of selected rows + scale
  {
    int rowsPerBlock = K6_WAVES * K6_ROWS_PER_WAVE;          // 32
    int blocks = (k + rowsPerBlock - 1) / rowsPerBlock;
    size_t shmem = (size_t)K6_WAVES * K6_ROWS_PER_WAVE * D * 4;  // 64KB @ D=512
    k6_gather<<<blocks, K6_WAVES * 32, shmem, stream>>>(x, sel, scl, outHidden, k, N, D);
  }

  (void)n_in; (void)out_size; (void)ws_size;
}